// LocalAggregator_46540265619955
// MI455X (gfx1250) — compile-verified
//
#include <hip/hip_runtime.h>
#include <hip/hip_bf16.h>

typedef __attribute__((ext_vector_type(2))) float v2f;
typedef __attribute__((ext_vector_type(8))) float v8f;

#define B_DIM 64
#define N_DIM 128
#define D_DIM 128
#define A_DIM 1000
#define KT    100   // K-panel for attr GEMM (1000 = 10 * 100)

__device__ __forceinline__ v8f wmma_f32(v2f a, v2f b, v8f c) {
  // V_WMMA_F32_16X16X4_F32 : exact fp32 matmul, matches reference precision
  return __builtin_amdgcn_wmma_f32_16x16x4_f32(false, a, false, b, (short)0, c,
                                               false, false);
}

// ---------------------------------------------------------------------------
// Kernel A: per (batch, 16-row i-tile) GAT attention.
//  e[i,j,k] = 0.6*(h_i .* a_k) . h_j + 0.4*(|h_i| .* a_k) . |h_j|
//  (leakyrelu(x*y) = 0.6*x*y + 0.4*|x||y| for slope 0.2)
// then adj-select, softmax over j, output = attn @ H.
// ---------------------------------------------------------------------------
__global__ __launch_bounds__(256) void gat_attn_kernel(
    const float* __restrict__ hidden, const int* __restrict__ adj,
    const float* __restrict__ amat, float* __restrict__ out) {
  __shared__ float sH[N_DIM * D_DIM];   // 64 KB: full H[b]
  __shared__ float sAlpha[16 * N_DIM];  // 8 KB: scores / attn for the i-tile
  __shared__ float sAa[D_DIM * 4];      // 2 KB: a[d][k]
  __shared__ float sRed[16][16];
  __shared__ float sMax[16];
  __shared__ float sSum[16];

  const int tid    = threadIdx.x;
  const int b      = blockIdx.x >> 3;
  const int i_base = (blockIdx.x & 7) * 16;

  // Stage H[b] and a into LDS with b128 loads.
  const float* Hg = hidden + (size_t)b * N_DIM * D_DIM;
  for (int idx4 = tid; idx4 < (N_DIM * D_DIM) / 4; idx4 += 256)
    reinterpret_cast<float4*>(sH)[idx4] = reinterpret_cast<const float4*>(Hg)[idx4];
  if (tid < (D_DIM * 4) / 4)
    reinterpret_cast<float4*>(sAa)[tid] = reinterpret_cast<const float4*>(amat)[tid];
  __syncthreads();

  const int wave   = tid >> 5;
  const int lane   = tid & 31;
  const int lo16   = lane & 15;
  const int hi16   = (lane >= 16) ? 8 : 0;   // C/D row offset for upper half-wave
  const int koff   = (lane >= 16) ? 2 : 0;   // A/B K offset for upper half-wave
  const int j_base = wave * 16;              // this wave's j-tile

  const float C1 = 0.6f, C2 = 0.4f;
  const float NEG_INF = -9.0e15f;

  // ---- score tiles for the 4 edge types, dual WMMA chains per K-step ----
  v8f acc[4];
  acc[0] = {}; acc[1] = {}; acc[2] = {}; acc[3] = {};
  const int i_row = i_base + lo16;
  const int j_row = j_base + lo16;
  for (int d0 = 0; d0 < D_DIM; d0 += 4) {
    const int d = d0 + koff;
    const float2 hi = *reinterpret_cast<const float2*>(&sH[i_row * D_DIM + d]);
    const float2 hj = *reinterpret_cast<const float2*>(&sH[j_row * D_DIM + d]);
    const float* a0 = &sAa[d * 4];
    const float* a1 = &sAa[(d + 1) * 4];
    const float hax = fabsf(hi.x), hay = fabsf(hi.y);
    v2f Bv;  Bv[0] = hj.x;        Bv[1] = hj.y;
    v2f Bva; Bva[0] = fabsf(hj.x); Bva[1] = fabsf(hj.y);
#pragma unroll
    for (int k = 0; k < 4; ++k) {
      v2f Av;  Av[0]  = C1 * hi.x * a0[k]; Av[1]  = C1 * hi.y * a1[k];
      acc[k] = wmma_f32(Av, Bv, acc[k]);
      v2f Ava; Ava[0] = C2 * hax * a0[k];  Ava[1] = C2 * hay * a1[k];
      acc[k] = wmma_f32(Ava, Bva, acc[k]);
    }
  }

  // ---- adjacency-based selection into alpha ----
  const int* adjB = adj + (size_t)b * N_DIM * N_DIM;
#pragma unroll
  for (int r = 0; r < 8; ++r) {
    const int il = r + hi16;
    const int t = adjB[(i_base + il) * N_DIM + j_base + lo16];
    float e = NEG_INF;
    e = (t == 1) ? acc[0][r] : e;
    e = (t == 2) ? acc[1][r] : e;
    e = (t == 3) ? acc[2][r] : e;
    e = (t == 4) ? acc[3][r] : e;
    sAlpha[il * N_DIM + j_base + lo16] = e;
  }
  __syncthreads();

  // ---- softmax over j (16 rows x 128 cols, 16 threads per row) ----
  const int row = tid >> 4;
  const int part = tid & 15;
  float m = -3.4e38f;
  for (int c = part * 8; c < part * 8 + 8; ++c)
    m = fmaxf(m, sAlpha[row * N_DIM + c]);
  sRed[row][part] = m;
  __syncthreads();
  if (part == 0) {
    float mm = sRed[row][0];
    for (int q = 1; q < 16; ++q) mm = fmaxf(mm, sRed[row][q]);
    sMax[row] = mm;
  }
  __syncthreads();
  const float mm2 = sMax[row];
  float s = 0.f;
  for (int c = part * 8; c < part * 8 + 8; ++c) {
    const float ev = __expf(sAlpha[row * N_DIM + c] - mm2);
    sAlpha[row * N_DIM + c] = ev;
    s += ev;
  }
  sRed[row][part] = s;
  __syncthreads();
  if (part == 0) {
    float ss = 0.f;
    for (int q = 0; q < 16; ++q) ss += sRed[row][q];
    sSum[row] = ss;
  }
  __syncthreads();
  const float inv = 1.0f / sSum[row];
  for (int c = part * 8; c < part * 8 + 8; ++c)
    sAlpha[row * N_DIM + c] *= inv;
  __syncthreads();

  // ---- output = attn[16x128] @ H[128x128]; wave owns a 16-wide d-tile ----
  const int dg = j_base + lo16;  // reuse wave tiling as d-column
  v8f oacc = {};
  for (int j0 = 0; j0 < N_DIM; j0 += 4) {
    const int jk = j0 + koff;
    const float2 av = *reinterpret_cast<const float2*>(&sAlpha[lo16 * N_DIM + jk]);
    v2f Av; Av[0] = av.x; Av[1] = av.y;
    v2f Bv; Bv[0] = sH[jk * D_DIM + dg]; Bv[1] = sH[(jk + 1) * D_DIM + dg];
    oacc = wmma_f32(Av, Bv, oacc);
  }
#pragma unroll
  for (int r = 0; r < 8; ++r)
    out[((size_t)b * N_DIM + i_base + r + hi16) * D_DIM + dg] = oacc[r];
}

// ---------------------------------------------------------------------------
// Kernel B: attr_sess = A_attr_sess[8192x1000] @ attr_embedding[1000x128]
// LDS-tiled fp32 WMMA GEMM, K-panels of 100.
// ---------------------------------------------------------------------------
__global__ __launch_bounds__(256) void attr_gemm_kernel(
    const float* __restrict__ Aattr, const float* __restrict__ Eemb,
    float* __restrict__ out2) {
  __shared__ float sAp[16 * KT];       // 6.25 KB
  __shared__ float sEp[KT * D_DIM];    // 50 KB

  const int tid    = threadIdx.x;
  const int m_base = blockIdx.x * 16;
  const int wave   = tid >> 5;
  const int lane   = tid & 31;
  const int lo16   = lane & 15;
  const int hi16   = (lane >= 16) ? 8 : 0;
  const int koff   = (lane >= 16) ? 2 : 0;
  const int dg     = wave * 16 + lo16;

  v8f acc = {};
  for (int kt = 0; kt < A_DIM / KT; ++kt) {
    const int k0 = kt * KT;
    __syncthreads();
    // A panel: 16 x 100 (b128 loads, rows 4000B apart -> 16B aligned)
    for (int idx4 = tid; idx4 < (16 * KT) / 4; idx4 += 256) {
      const int r = idx4 / (KT / 4), c4 = idx4 % (KT / 4);
      *reinterpret_cast<float4*>(&sAp[r * KT + c4 * 4]) =
          *reinterpret_cast<const float4*>(
              &Aattr[(size_t)(m_base + r) * A_DIM + k0 + c4 * 4]);
    }
    // B panel: 100 x 128 (b128 loads)
    for (int idx4 = tid; idx4 < (KT * D_DIM) / 4; idx4 += 256) {
      const int r = idx4 >> 5, c4 = idx4 & 31;
      *reinterpret_cast<float4*>(&sEp[r * D_DIM + c4 * 4]) =
          *reinterpret_cast<const float4*>(
              &Eemb[(size_t)(k0 + r) * D_DIM + c4 * 4]);
    }
    __syncthreads();
    if (kt + 1 < A_DIM / KT)
      __builtin_prefetch(&Aattr[(size_t)m_base * A_DIM + k0 + KT], 0, 0);
#pragma unroll
    for (int ks = 0; ks < KT / 4; ++ks) {
      const int k = 4 * ks + koff;
      const float2 av = *reinterpret_cast<const float2*>(&sAp[lo16 * KT + k]);
      v2f Av; Av[0] = av.x; Av[1] = av.y;
      v2f Bv; Bv[0] = sEp[k * D_DIM + dg]; Bv[1] = sEp[(k + 1) * D_DIM + dg];
      acc = wmma_f32(Av, Bv, acc);
    }
  }
#pragma unroll
  for (int r = 0; r < 8; ++r)
    out2[(size_t)(m_base + r + hi16) * D_DIM + dg] = acc[r];
}

extern "C" void kernel_launch(void* const* d_in, const int* in_sizes, int n_in,
                              void* d_out, int out_size, void* d_ws, size_t ws_size,
                              hipStream_t stream) {
  (void)in_sizes; (void)n_in; (void)out_size; (void)d_ws; (void)ws_size;
  const float* hidden = (const float*)d_in[0];
  const int*   adj    = (const int*)d_in[1];
  const float* amat   = (const float*)d_in[2];
  const float* Aattr  = (const float*)d_in[3];
  const float* Eemb   = (const float*)d_in[4];

  float* out_attn = (float*)d_out;                                    // [B,N,D]
  float* out_attr = out_attn + (size_t)B_DIM * N_DIM * D_DIM;         // [B,N,D]

  gat_attn_kernel<<<dim3(B_DIM * (N_DIM / 16)), dim3(256), 0, stream>>>(
      hidden, adj, amat, out_attn);
  attr_gemm_kernel<<<dim3((B_DIM * N_DIM) / 16), dim3(256), 0, stream>>>(
      Aattr, Eemb, out_attr);
}